// _MambaState_73632919322669
// MI455X (gfx1250) — compile-verified
//
#include <hip/hip_runtime.h>

// Mamba-style fused state update, f32, [8192 x 4096].
// Pure HBM-bandwidth-bound elementwise map (~768 MiB moved, ~0.17 GFLOP):
// roofline floor ~34us at 23.3 TB/s. WMMA is irrelevant here; the win is
// wave32 b128 streaming with non-temporal hints so the one-touch stream
// doesn't thrash the 192MB L2, and per-column sigmoid/softplus hoisted out
// of the row loop.

typedef float __attribute__((ext_vector_type(4))) fv4;

#define BATCH 8192
#define DIM   4096
#define TPB   256                       // 8 waves (wave32) per block
#define COLS_PER_BLOCK (TPB * 4)        // 1024 columns per block (float4/thread)
#define GRID_X (DIM / COLS_PER_BLOCK)   // 4 column blocks
#define GRID_Y 2048                     // 8192/2048 = 4 rows per thread, exact

__global__ __launch_bounds__(TPB) void _MambaState_kernel(
    const float* __restrict__ prev,
    const float* __restrict__ b_t,
    const float* __restrict__ v_t,
    const float* __restrict__ c_t,
    const float* __restrict__ a_logit,
    const float* __restrict__ delta_log,
    float* __restrict__ out_state,
    float* __restrict__ out_y)
{
    const int col = (blockIdx.x * TPB + threadIdx.x) * 4;   // 4 consecutive cols

    // Per-column params: tiny [4096] vectors, reused by all 2048 row-blocks ->
    // regular (RT) loads so they stay resident in L2.
    const fv4 al = *(const fv4*)(a_logit  + col);
    const fv4 dl = *(const fv4*)(delta_log + col);

    // Hoist the only transcendentals out of the streaming loop:
    // a = sigmoid(a_logit), dt = softplus(delta_log), computed once per thread.
    fv4 a, dt;
#pragma unroll
    for (int i = 0; i < 4; ++i) {
        a[i] = 1.0f / (1.0f + __expf(-al[i]));
        const float x = dl[i];
        dt[i] = (x > 20.0f) ? x : __logf(1.0f + __expf(x));  // softplus, overflow-safe
    }

    // Grid-stride over rows: 4 iterations exactly, no divergence.
    for (int row = blockIdx.y; row < BATCH; row += GRID_Y) {
        const size_t base = (size_t)row * DIM + col;

        // One-touch 512MiB input stream: non-temporal b128 loads.
        const fv4 p = __builtin_nontemporal_load((const fv4*)(prev + base));
        const fv4 b = __builtin_nontemporal_load((const fv4*)(b_t  + base));
        const fv4 v = __builtin_nontemporal_load((const fv4*)(v_t  + base));
        const fv4 c = __builtin_nontemporal_load((const fv4*)(c_t  + base));

        const fv4 ns = a * p + dt * (b * v);   // v_fma / v_mul packed f32
        const fv4 yy = c * ns;

        // Write-only 256MiB output stream: non-temporal b128 stores
        // (fully coalesced: 256 threads x 16B = 4KiB contiguous per block-row).
        __builtin_nontemporal_store(ns, (fv4*)(out_state + base));
        __builtin_nontemporal_store(yy, (fv4*)(out_y     + base));
    }
}

extern "C" void kernel_launch(void* const* d_in, const int* in_sizes, int n_in,
                              void* d_out, int out_size, void* d_ws, size_t ws_size,
                              hipStream_t stream) {
    const float* prev      = (const float*)d_in[0];
    const float* b_t       = (const float*)d_in[1];
    const float* v_t       = (const float*)d_in[2];
    const float* c_t       = (const float*)d_in[3];
    const float* a_logit   = (const float*)d_in[4];
    const float* delta_log = (const float*)d_in[5];

    // Tuple output (new_state, y) concatenated flat.
    float* out_state = (float*)d_out;
    float* out_y     = (float*)d_out + (size_t)BATCH * (size_t)DIM;

    dim3 grid(GRID_X, GRID_Y, 1);
    _MambaState_kernel<<<grid, dim3(TPB, 1, 1), 0, stream>>>(
        prev, b_t, v_t, c_t, a_logit, delta_log, out_state, out_y);
}